// ReactantStage2_26723286516090
// MI455X (gfx1250) — compile-verified
//
#include <hip/hip_runtime.h>
#include <hip/hip_bf16.h>

// ---------------------------------------------------------------------------
// Sizes (from reference): N=100000 nodes, E=1600000 edges, D=128, G=256 graphs
// Output: [N, 2D] fp32.
// ---------------------------------------------------------------------------

#define D 128
#define NGRAPHS 256
#define NTILES 8            // D/16 column tiles
#define NKSTEPS 32          // D/4 k-steps
#define WPACK_ELEMS (NTILES * NKSTEPS * 32 * 2)   // 16384 floats = 64 KB

typedef float v2f __attribute__((ext_vector_type(2)));
typedef float v8f __attribute__((ext_vector_type(8)));

// ---------------------------- generic zero fill ----------------------------
__global__ void zero_f32(float* __restrict__ p, long n) {
  long i = (long)blockIdx.x * blockDim.x + threadIdx.x;
  long stride = (long)gridDim.x * blockDim.x;
  for (; i < n; i += stride) p[i] = 0.0f;
}

// -------------------- pack weights into WMMA fragment order ----------------
// Wp[((tile*32 + ks)*32 + lane)*2 + j] = W[(4*ks + 2*(lane>>4) + j)*D + tile*16 + (lane&15)]
// so each wave's per-step B fragment is one coalesced global_load_b64 per lane.
__global__ __launch_bounds__(256) void pack_weights(const float* __restrict__ W,
                                                    float* __restrict__ Wp) {
  int idx = blockIdx.x * blockDim.x + threadIdx.x;
  if (idx >= WPACK_ELEMS) return;
  int j    = idx & 1;
  int lane = (idx >> 1) & 31;
  int ks   = (idx >> 6) & 31;
  int tile = idx >> 11;
  int k = 4 * ks + 2 * (lane >> 4) + j;
  int c = tile * 16 + (lane & 15);
  Wp[idx] = W[k * D + c];
}

// ---------------------- edge gather + scatter-add (agg) --------------------
// One wave per edge; each lane handles 4 consecutive floats of the 128 row.
__global__ __launch_bounds__(256) void edge_scatter(
    const int* __restrict__ ei, const float* __restrict__ x,
    const float* __restrict__ ea, float* __restrict__ agg, int E) {
  long gid = (long)blockIdx.x * blockDim.x + threadIdx.x;
  long e = gid >> 5;
  if (e >= E) return;
  int lane = (int)(gid & 31);
  int c = lane * 4;
  int s = ei[e];           // edge_index[0][e]
  int d = ei[(long)E + e]; // edge_index[1][e]
  const float4 xv = *(const float4*)(x + (long)s * D + c);
  const float4 ev = *(const float4*)(ea + e * (long)D + c);
  float* p = agg + (long)d * D + c;
  unsafeAtomicAdd(p + 0, xv.x + ev.x);
  unsafeAtomicAdd(p + 1, xv.y + ev.y);
  unsafeAtomicAdd(p + 2, xv.z + ev.z);
  unsafeAtomicAdd(p + 3, xv.w + ev.w);
}

// ------------------------- per-graph node counts ---------------------------
__global__ void gcount_kernel(const int* __restrict__ batch, int* __restrict__ gcnt, int n) {
  int i = blockIdx.x * blockDim.x + threadIdx.x;
  if (i < n) atomicAdd(&gcnt[batch[i]], 1);
}

__global__ void gscan_kernel(const int* __restrict__ gcnt, int* __restrict__ gstart) {
  if (blockIdx.x == 0 && threadIdx.x == 0) {
    int acc = 0;
    for (int g = 0; g < NGRAPHS; ++g) { gstart[g] = acc; acc += gcnt[g]; }
    gstart[NGRAPHS] = acc;
  }
}

// --------------------- fused 2-layer MLP via fp32 WMMA ---------------------
// Block: 256 threads = 8 waves; handles 16 rows. Wave w owns output columns
// [16w, 16w+16). h = relu((x+agg)@W1 + b1)@W2 + b2, written over agg buffer.
__global__ __launch_bounds__(256) void mlp_wmma(
    const float* __restrict__ x, float* __restrict__ hb /* in: agg, out: h */,
    const float* __restrict__ Wp1, const float* __restrict__ b1v,
    const float* __restrict__ Wp2, const float* __restrict__ b2v, int n) {
  __shared__ float xin[16][D];
  __shared__ float ts[16][D];
  const int tid = threadIdx.x;
  const int lane = tid & 31;
  const int wv = tid >> 5;
  const int half = lane >> 4;   // 0: lanes 0-15, 1: lanes 16-31
  const int l15 = lane & 15;
  const int cn = wv * 16 + l15; // output column owned by this lane's tile
  const long row0 = (long)blockIdx.x * 16;

  // stage in xin = x + agg (16 x 128), float4-vectorized
  for (int i = tid; i < 16 * (D / 4); i += 256) {
    int r = i >> 5, c4 = (i & 31) * 4;
    long gr = row0 + r;
    float4 v = make_float4(0.f, 0.f, 0.f, 0.f);
    if (gr < n) {
      float4 a4 = *(const float4*)(x + gr * D + c4);
      float4 g4 = *(const float4*)(hb + gr * D + c4);
      v = make_float4(a4.x + g4.x, a4.y + g4.y, a4.z + g4.z, a4.w + g4.w);
    }
    *(float4*)&xin[r][c4] = v;
  }
  __syncthreads();

  const float* frag1 = Wp1 + (size_t)wv * NKSTEPS * 64 + lane * 2;
  const float* frag2 = Wp2 + (size_t)wv * NKSTEPS * 64 + lane * 2;

  // ---- layer 1: t = relu(xin @ W1 + b1) ----
  v8f acc = {};
#pragma unroll
  for (int ks = 0; ks < NKSTEPS; ++ks) {
    const int ka = 4 * ks + 2 * half;
    v2f a, b;
    a.x = xin[l15][ka];
    a.y = xin[l15][ka + 1];
    b = *(const v2f*)(frag1 + ks * 64);
    acc = __builtin_amdgcn_wmma_f32_16x16x4_f32(false, a, false, b,
                                                (short)0, acc, false, false);
  }
  {
    const float bb = b1v[cn];
#pragma unroll
    for (int r = 0; r < 8; ++r) {
      float v = acc[r] + bb;
      ts[r + 8 * half][cn] = v > 0.0f ? v : 0.0f;
    }
  }
  __syncthreads();

  // ---- layer 2: h = t @ W2 + b2 ----
  v8f acc2 = {};
#pragma unroll
  for (int ks = 0; ks < NKSTEPS; ++ks) {
    const int ka = 4 * ks + 2 * half;
    v2f a, b;
    a.x = ts[l15][ka];
    a.y = ts[l15][ka + 1];
    b = *(const v2f*)(frag2 + ks * 64);
    acc2 = __builtin_amdgcn_wmma_f32_16x16x4_f32(false, a, false, b,
                                                 (short)0, acc2, false, false);
  }
  {
    const float bb = b2v[cn];
#pragma unroll
    for (int r = 0; r < 8; ++r) {
      long gr = row0 + r + 8 * half;
      if (gr < n) hb[gr * D + cn] = acc2[r] + bb;
    }
  }
}

// ------------- exclusive prefix sum of primary flag over nodes -------------
__global__ __launch_bounds__(256) void scan_local(
    const int* __restrict__ primary, int* __restrict__ P,
    int* __restrict__ bsums, int n) {
  __shared__ int sm[256];
  const int t = threadIdx.x;
  const int i = blockIdx.x * 256 + t;
  const int f = (i < n && primary[i] == 1) ? 1 : 0;
  sm[t] = f;
  __syncthreads();
  for (int off = 1; off < 256; off <<= 1) {
    int v = (t >= off) ? sm[t - off] : 0;
    __syncthreads();
    sm[t] += v;
    __syncthreads();
  }
  if (i < n) P[i] = sm[t] - f;   // local exclusive prefix
  if (t == 255) bsums[blockIdx.x] = sm[255];
}

__global__ void scan_bsums(int* __restrict__ bsums, int nb) {
  if (blockIdx.x == 0 && threadIdx.x == 0) {
    int acc = 0;
    for (int i = 0; i < nb; ++i) { int v = bsums[i]; bsums[i] = acc; acc += v; }
  }
}

__global__ __launch_bounds__(256) void scan_add(int* __restrict__ P,
                                                const int* __restrict__ bsums, int n) {
  int i = blockIdx.x * 256 + threadIdx.x;
  if (i < n) P[i] += bsums[blockIdx.x];
}

// -------------------- condition-node pooling accumulation ------------------
__global__ __launch_bounds__(128) void pool_acc(
    const float* __restrict__ h, const int* __restrict__ batch,
    const int* __restrict__ primary, float* __restrict__ csum,
    float* __restrict__ ccnt, int n) {
  const int node = blockIdx.x;
  if (node >= n || primary[node] != 0) return;
  const int g = batch[node];
  const int d = threadIdx.x;
  unsafeAtomicAdd(&csum[g * D + d], h[(long)node * D + d]);
  if (d == 0) unsafeAtomicAdd(&ccnt[g], 1.0f);
}

// ------------------------------- assembly ----------------------------------
// out pre-zeroed; write only primary rows [h | pool] at their sorted position.
__global__ __launch_bounds__(256) void assemble(
    const float* __restrict__ h, const int* __restrict__ batch,
    const int* __restrict__ primary, const int* __restrict__ gstart,
    const int* __restrict__ P, const float* __restrict__ csum,
    const float* __restrict__ ccnt, float* __restrict__ out, int n) {
  const int node = blockIdx.x;
  if (node >= n || primary[node] != 1) return;
  const int g = batch[node];
  const int pos = gstart[g] + P[node] - P[gstart[g]];
  const int t = threadIdx.x;
  const long ob = (long)pos * (2 * D);
  if (t < D) {
    out[ob + t] = h[(long)node * D + t];
  } else {
    const int d = t - D;
    const float cnt = ccnt[g];
    out[ob + D + d] = cnt > 0.0f ? csum[g * D + d] / cnt : 0.0f;
  }
}

// ---------------------------------------------------------------------------
extern "C" void kernel_launch(void* const* d_in, const int* in_sizes, int n_in,
                              void* d_out, int out_size, void* d_ws, size_t ws_size,
                              hipStream_t stream) {
  const float* x       = (const float*)d_in[0];
  const int*   ei      = (const int*)  d_in[1];
  const float* ea      = (const float*)d_in[2];
  const int*   batch   = (const int*)  d_in[3];
  const int*   primary = (const int*)  d_in[4];
  const float* W1      = (const float*)d_in[5];
  const float* b1v     = (const float*)d_in[6];
  const float* W2      = (const float*)d_in[7];
  const float* b2v     = (const float*)d_in[8];

  const int N = in_sizes[3];
  const int E = in_sizes[2] / D;
  const int nb = (N + 255) / 256;

  // workspace carve-up (16B-aligned sections first)
  char* ws = (char*)d_ws;
  float* agg   = (float*)ws; ws += (size_t)N * D * sizeof(float);   // reused as h
  float* Wp1   = (float*)ws; ws += (size_t)WPACK_ELEMS * sizeof(float);
  float* Wp2   = (float*)ws; ws += (size_t)WPACK_ELEMS * sizeof(float);
  float* csum  = (float*)ws; ws += (size_t)NGRAPHS * D * sizeof(float);
  float* ccnt  = (float*)ws; ws += (size_t)NGRAPHS * sizeof(float);
  int*   gcnt  = (int*)ws;   ws += (size_t)NGRAPHS * sizeof(int);
  int*   gstart= (int*)ws;   ws += (size_t)(NGRAPHS + 1) * sizeof(int);
  int*   P     = (int*)ws;   ws += (size_t)N * sizeof(int);
  int*   bsums = (int*)ws;   ws += (size_t)nb * sizeof(int);

  // 1) zero accumulation buffers + output; pack weights into fragment order
  zero_f32<<<4096, 256, 0, stream>>>(agg, (long)N * D);
  zero_f32<<<64, 256, 0, stream>>>(csum, (long)NGRAPHS * D + NGRAPHS + NGRAPHS);
  zero_f32<<<4096, 256, 0, stream>>>((float*)d_out, (long)out_size);
  pack_weights<<<(WPACK_ELEMS + 255) / 256, 256, 0, stream>>>(W1, Wp1);
  pack_weights<<<(WPACK_ELEMS + 255) / 256, 256, 0, stream>>>(W2, Wp2);

  // 2) edge gather + scatter-add
  {
    long threads = (long)E * 32;
    int blocks = (int)((threads + 255) / 256);
    edge_scatter<<<blocks, 256, 0, stream>>>(ei, x, ea, agg, E);
  }

  // 3) per-graph counts + exclusive scan -> graph starts
  gcount_kernel<<<nb, 256, 0, stream>>>(batch, gcnt, N);
  gscan_kernel<<<1, 32, 0, stream>>>(gcnt, gstart);

  // 4) fused 2-layer MLP with fp32 WMMA (h overwrites agg)
  {
    int blocks = (N + 15) / 16;
    mlp_wmma<<<blocks, 256, 0, stream>>>(x, agg, Wp1, b1v, Wp2, b2v, N);
  }

  // 5) exclusive prefix sum of primary flags
  scan_local<<<nb, 256, 0, stream>>>(primary, P, bsums, N);
  scan_bsums<<<1, 32, 0, stream>>>(bsums, nb);
  scan_add<<<nb, 256, 0, stream>>>(P, bsums, N);

  // 6) condition-node pooling
  pool_acc<<<N, 128, 0, stream>>>(agg, batch, primary, csum, ccnt, N);

  // 7) assemble output rows
  assemble<<<N, 256, 0, stream>>>(agg, batch, primary, gstart, P, csum, ccnt,
                                  (float*)d_out, N);
}